// LSTM_41077067219711
// MI455X (gfx1250) — compile-verified
//
#include <hip/hip_runtime.h>
#include <math.h>

// LSTM (B=32, T=2048, I=H=512) for MI455X / gfx1250.
// Persistent-grid recurrence with per-step device-wide barrier.
// GEMMs use V_WMMA_F32_16X16X4_F32 (full fp32 precision, matches reference).

typedef __attribute__((ext_vector_type(2))) float v2f;
typedef __attribute__((ext_vector_type(8))) float v8f;

#define BB 32
#define TT 2048
#define II 512
#define HH 512
#define NWG 32        // 32 persistent workgroups, each owns 16 hidden channels
#define NTHREADS 256  // 8 waves: wave = (gate<<1)|mtile -> one 16x16 z-tile each

__device__ __forceinline__ float fsig(float x) {
  return 1.0f / (1.0f + __expf(-x));
}
__device__ __forceinline__ float ftanh(float x) {
  // tanh(x) = 2*sigmoid(2x) - 1, via hw v_exp_f32
  return 2.0f * fsig(2.0f * x) - 1.0f;
}

__device__ __forceinline__ void gridBarrier(unsigned* cnt, unsigned* gen,
                                            unsigned expect) {
  __syncthreads();
  if (threadIdx.x == 0) {
    __threadfence();
    unsigned old = atomicAdd(cnt, 1u);
    if (old == (unsigned)(NWG - 1)) {
      atomicExch(cnt, 0u);     // reset for next round (safe: nobody re-arrives
      __threadfence();         // until they observe gen advance)
      atomicAdd(gen, 1u);      // release
    } else {
      while (__hip_atomic_load(gen, __ATOMIC_ACQUIRE,
                               __HIP_MEMORY_SCOPE_AGENT) < expect) {
        __builtin_amdgcn_s_sleep(2);
      }
    }
    __threadfence();
  }
  __syncthreads();
}

__global__ void lstm_init_ws(unsigned* cnt, unsigned* gen) {
  *cnt = 0u;
  *gen = 0u;
}

__global__ __launch_bounds__(NTHREADS) void lstm_persistent(
    const float* __restrict__ x,    // [B,T,I]
    const float* __restrict__ Wih,  // [4H,I] row-major
    const float* __restrict__ Whh,  // [4H,H] row-major
    const float* __restrict__ bih,  // [4H]
    const float* __restrict__ bhh,  // [4H]
    float* __restrict__ out,        // [1,B,H]
    float* __restrict__ h0buf,      // [B,H] ping
    float* __restrict__ h1buf,      // [B,H] pong
    unsigned* cnt, unsigned* gen) {
  __shared__ float sz[4][BB][16];   // z-tiles, gate-major     (8 KB)
  __shared__ float sc[BB][16];      // cell state, WG-private  (2 KB)
  __shared__ float sbias[4][16];    // b_ih + b_hh slice

  const int tid = threadIdx.x;
  const int wg = blockIdx.x;        // channel chunk: cols wg*16 .. wg*16+15
  const int wave = tid >> 5;
  const int lane = tid & 31;
  const int gate = wave >> 1;       // 0=i 1=f 2=g 3=o
  const int mtile = wave & 1;       // batch rows 0-15 or 16-31
  const int l16 = lane & 15;
  const int ksel = (lane >> 4) << 1;         // K-pair select: 0 or 2
  const int m = mtile * 16 + l16;            // batch row (A fragment row)
  const int n = gate * HH + wg * 16 + l16;   // weight row (B fragment col)

  // --- init: zero this WG's slice of h0, zero c, stage bias ---
  for (int e = tid; e < BB * 16; e += NTHREADS) {
    int b = e >> 4, j = e & 15;
    h0buf[b * HH + wg * 16 + j] = 0.0f;
    sc[b][j] = 0.0f;
  }
  if (tid < 64) {
    int g = tid >> 4, j = tid & 15;
    int col = g * HH + wg * 16 + j;
    sbias[g][j] = bih[col] + bhh[col];
  }

  unsigned barNum = 0;
  gridBarrier(cnt, gen, ++barNum);  // h0 zeros visible device-wide

  for (int t = 0; t < TT; ++t) {
    const float* __restrict__ hcur = (t & 1) ? h1buf : h0buf;
    float* __restrict__ hnext = (t & 1) ? h0buf : h1buf;

    const float* __restrict__ hrow = hcur + m * HH + ksel;
    const float* __restrict__ xrow = x + ((size_t)m * TT + t) * II + ksel;
    const float* __restrict__ whh_row = Whh + (size_t)n * HH + ksel;
    const float* __restrict__ wih_row = Wih + (size_t)n * II + ksel;

    v8f acc = {};
    // z += h @ W_hh^T   (K = 512, 128 wmma steps)
#pragma unroll 8
    for (int kk = 0; kk < HH; kk += 4) {
      v2f a = *(const v2f*)(hrow + kk);
      v2f b = *(const v2f*)(whh_row + kk);
      acc = __builtin_amdgcn_wmma_f32_16x16x4_f32(false, a, false, b,
                                                  (short)0, acc, false, false);
    }
    // z += x_t @ W_ih^T (K = 512, 128 wmma steps)
#pragma unroll 8
    for (int kk = 0; kk < II; kk += 4) {
      v2f a = *(const v2f*)(xrow + kk);
      v2f b = *(const v2f*)(wih_row + kk);
      acc = __builtin_amdgcn_wmma_f32_16x16x4_f32(false, a, false, b,
                                                  (short)0, acc, false, false);
    }
    // warm L2 for next timestep's x slice (global_prefetch_b8)
    if (t + 1 < TT) __builtin_prefetch(xrow + II, 0, 1);

    // scatter 16x16 f32 D-tile to LDS per documented C/D layout:
    // VGPR r -> M = r + 8*(lane/16), N = lane%16
    {
      const int mr = (lane >> 4) * 8;
#pragma unroll
      for (int r = 0; r < 8; ++r)
        sz[gate][mtile * 16 + mr + r][l16] = acc[r];
    }
    __syncthreads();

    // pointwise gates: 512 (b,j) pairs, 2 per thread
#pragma unroll
    for (int e = tid; e < BB * 16; e += NTHREADS) {
      int b = e >> 4, j = e & 15;
      float zi = sz[0][b][j] + sbias[0][j];
      float zf = sz[1][b][j] + sbias[1][j];
      float zg = sz[2][b][j] + sbias[2][j];
      float zo = sz[3][b][j] + sbias[3][j];
      float ig = fsig(zi);
      float fg = fsig(zf);
      float gg = ftanh(zg);
      float og = fsig(zo);
      float cn = fg * sc[b][j] + ig * gg;
      sc[b][j] = cn;
      float hn = og * ftanh(cn);
      hnext[b * HH + wg * 16 + j] = hn;
      if (t == TT - 1) out[b * HH + wg * 16 + j] = hn;
    }

    gridBarrier(cnt, gen, ++barNum);  // publish h for step t+1
  }
}

extern "C" void kernel_launch(void* const* d_in, const int* in_sizes, int n_in,
                              void* d_out, int out_size, void* d_ws,
                              size_t ws_size, hipStream_t stream) {
  const float* x = (const float*)d_in[0];
  const float* Wih = (const float*)d_in[1];
  const float* Whh = (const float*)d_in[2];
  const float* bih = (const float*)d_in[3];
  const float* bhh = (const float*)d_in[4];
  float* out = (float*)d_out;

  unsigned char* ws = (unsigned char*)d_ws;
  unsigned* cnt = (unsigned*)ws;
  unsigned* gen = (unsigned*)(ws + 64);
  float* h0 = (float*)(ws + 256);
  float* h1 = (float*)(ws + 256 + (size_t)BB * HH * sizeof(float));

  hipLaunchKernelGGL(lstm_init_ws, dim3(1), dim3(1), 0, stream, cnt, gen);
  hipLaunchKernelGGL(lstm_persistent, dim3(NWG), dim3(NTHREADS), 0, stream,
                     x, Wih, Whh, bih, bhh, out, h0, h1, cnt, gen);
}